// CTCWithoutBlankLoss_1116691497598
// MI455X (gfx1250) — compile-verified
//
#include <hip/hip_runtime.h>
#include <hip/hip_bf16.h>
#include <stdint.h>

// CTC-without-blank loss, fused for MI455X (gfx1250).
// B=16, T=1024, V=4096, S=256.
namespace {
constexpr int   kB     = 16;
constexpr int   kT     = 1024;
constexpr int   kV     = 4096;
constexpr int   kS     = 256;
constexpr float kNeg   = -1e30f;          // log-domain zero sentinel
constexpr float kLog2e = 1.44269504088896340736f;
constexpr float kLn2   = 0.69314718055994530942f;
}

// Raw hardware base-2 transcendentals (v_exp_f32 / v_log_f32).
__device__ __forceinline__ float hw_exp2(float x) { return __builtin_amdgcn_exp2f(x); }
__device__ __forceinline__ float hw_log2(float x) { return __builtin_amdgcn_logf(x); }

// ---------------------------------------------------------------------------
// Phase 1: one block per (b,t) row. Async-DMA the 16KB row into LDS (NT hint:
// logits are streamed once and must not evict the lp workspace from L2),
// compute logsumexp, then gather the S target columns. Output is pre-scaled
// to the log2 domain:  lp2[b,t,s] = (logits[b,t,tgt] - lse(b,t)) * log2(e)
// ---------------------------------------------------------------------------
__global__ __launch_bounds__(256) void ctc_row_lse_gather(
    const float* __restrict__ logits,
    const int*   __restrict__ targets,
    float*       __restrict__ lp2)
{
  __shared__ float rowbuf[kV];   // 16 KB staged row
  __shared__ float sw[16];       // [0..7] per-wave maxes, [8..15] per-wave sums

  const int row  = blockIdx.x;         // row = b*kT + t
  const int b    = row >> 10;          // / kT
  const int tid  = threadIdx.x;
  const int lane = tid & 31;
  const int w    = tid >> 5;

  // --- async global -> LDS copy (ASYNCcnt-tracked DMA, non-temporal) ---
  // 4 passes x 256 lanes x 16B = 16 KB.
  const uint64_t gbase    = (uint64_t)(uintptr_t)logits + (uint64_t)row * (kV * 4);
  const uint32_t lds_base = (uint32_t)(uintptr_t)(&rowbuf[0]);
  #pragma unroll
  for (int p = 0; p < 4; ++p) {
    uint32_t voff = (uint32_t)(tid * 16 + p * 4096);
    uint32_t la   = lds_base + voff;
    asm volatile("global_load_async_to_lds_b128 %0, %1, %2 th:TH_LOAD_NT"
                 :: "v"(la), "v"(voff), "s"(gbase)
                 : "memory");
  }
  asm volatile("s_wait_asynccnt 0" ::: "memory");
  __syncthreads();

  // --- pull this thread's 16 strided elements into registers (read LDS once)
  float x[16];
  #pragma unroll
  for (int k = 0; k < 16; ++k) x[k] = rowbuf[tid + (k << 8)];

  // --- row max: register tree + wave32 shfl_xor tree + 8-slot LDS exchange
  float m = x[0];
  #pragma unroll
  for (int k = 1; k < 16; ++k) m = fmaxf(m, x[k]);
  #pragma unroll
  for (int off = 16; off > 0; off >>= 1) m = fmaxf(m, __shfl_xor(m, off, 32));
  if (lane == 0) sw[w] = m;
  __syncthreads();
  float rowmax = sw[0];
  #pragma unroll
  for (int i = 1; i < 8; ++i) rowmax = fmaxf(rowmax, sw[i]);

  // --- sum of exp (base-2 hardware transcendentals) ---
  float ssum = 0.0f;
  #pragma unroll
  for (int k = 0; k < 16; ++k) ssum += hw_exp2((x[k] - rowmax) * kLog2e);
  #pragma unroll
  for (int off = 16; off > 0; off >>= 1) ssum += __shfl_xor(ssum, off, 32);
  if (lane == 0) sw[8 + w] = ssum;
  __syncthreads();
  float tot = sw[8];
  #pragma unroll
  for (int i = 1; i < 8; ++i) tot += sw[8 + i];

  const float l2tot = hw_log2(tot);

  // --- gather: thread tid == target position s; emit log2-domain lp ---
  const int tgt = targets[b * kS + tid];
  lp2[(size_t)row * kS + tid] = (rowbuf[tgt] - rowmax) * kLog2e - l2tot;
}

// ---------------------------------------------------------------------------
// Phase 2: one block per batch b; thread s holds beta[s] = alpha[s]*log2(e)
// in a register. Recurrence in the log2 domain (v_exp_f32/v_log_f32 native):
//   beta[s] = lp2[t,s] + max + log2(1 + exp2(min - max))
// Intra-wave shift via shfl_up; wave boundaries via parity-buffered LDS
// (one barrier per timestep). lp2 for t+1 prefetched one step ahead (L2 hit).
// ---------------------------------------------------------------------------
__global__ __launch_bounds__(256) void ctc_scan(
    const float* __restrict__ lp2,
    const int*   __restrict__ logits_lengths,
    const int*   __restrict__ targets_lengths,
    float*       __restrict__ loss_b)
{
  __shared__ float bnd[2][8];          // lane-31 beta of each wave, per parity
  const int b    = blockIdx.x;
  const int s    = threadIdx.x;
  const int lane = s & 31;
  const int w    = s >> 5;
  const int TL   = logits_lengths[b];
  const float* lpb = lp2 + (size_t)b * kT * kS;

  float beta = kNeg;
  if (s < 8) bnd[0][s] = kNeg;
  __syncthreads();

  float lv = lpb[s];                   // lp2 at t = 0
  for (int t = 0; t < TL; ++t) {
    // prefetch next timestep (hides L2 latency under compute + barrier)
    float lv_next = (t + 1 < TL) ? lpb[(size_t)(t + 1) * kS + s] : 0.0f;

    const int p = t & 1;
    float up = __shfl_up(beta, 1, 32);
    float shifted;
    if (lane == 0) {
      shifted = (w == 0) ? ((t == 0) ? 0.0f : kNeg) : bnd[p][w - 1];
    } else {
      shifted = up;
    }
    // log2-domain logaddexp, NaN-free for kNeg inputs
    float mx = fmaxf(beta, shifted);
    float mn = fminf(beta, shifted);
    beta = lv + mx + hw_log2(1.0f + hw_exp2(mn - mx));

    if (lane == 31) bnd[p ^ 1][w] = beta;
    __syncthreads();
    lv = lv_next;
  }

  if (s == targets_lengths[b] - 1) loss_b[b] = -beta;   // still log2-domain
}

// ---------------------------------------------------------------------------
// Phase 3: deterministic scalar reduction; convert log2 domain back (x ln2).
// ---------------------------------------------------------------------------
__global__ void ctc_reduce(const float* __restrict__ loss_b,
                           float* __restrict__ out)
{
  if (threadIdx.x == 0 && blockIdx.x == 0) {
    float acc = 0.0f;
    #pragma unroll
    for (int b = 0; b < kB; ++b) acc += loss_b[b];
    out[0] = acc * kLn2;
  }
}

extern "C" void kernel_launch(void* const* d_in, const int* in_sizes, int n_in,
                              void* d_out, int out_size, void* d_ws, size_t ws_size,
                              hipStream_t stream) {
  (void)in_sizes; (void)n_in; (void)out_size; (void)ws_size;
  const float* logits          = (const float*)d_in[0];
  const int*   targets         = (const int*)d_in[1];
  const int*   logits_lengths  = (const int*)d_in[2];
  const int*   targets_lengths = (const int*)d_in[3];

  // workspace layout: lp2[B*T*S] floats (16 MB), then loss_b[B] floats
  float* lp2    = (float*)d_ws;
  float* loss_b = lp2 + (size_t)kB * kT * kS;

  ctc_row_lse_gather<<<kB * kT, 256, 0, stream>>>(logits, targets, lp2);
  ctc_scan<<<kB, 256, 0, stream>>>(lp2, logits_lengths, targets_lengths, loss_b);
  ctc_reduce<<<1, 32, 0, stream>>>(loss_b, (float*)d_out);
}